// QuadraticChargesEnergyReadout_13872744366321
// MI455X (gfx1250) — compile-verified
//
#include <hip/hip_runtime.h>

// ---------------------------------------------------------------------------
// MI455X (gfx1250): quadratic charges energy readout via V_WMMA_F32_16X16X4_F32.
//
//  - symmetric-packed W_ss/W_vv, row-padded to K=576: branch-free K-sweep
//  - merged W_sv/W_vs (K=1024, u-major rows)
//  - vv / qv passes process all 3 vector components per step, sharing the
//    B fragment; qv additionally preloads its 24 y-pairs into registers
//    (column index independent of row) -> 2 DS loads per 6 WMMAs
//  - all quadratic weights staged in LDS (~278KB of 320KB/WGP)
//  - readout GEMM (Wd^T @ nf) in the same WMMA shape -> energy is an
//    elementwise acc*t + 16-lane LDS reduction
// ---------------------------------------------------------------------------

typedef __attribute__((ext_vector_type(2))) float v2f;
typedef __attribute__((ext_vector_type(8))) float v8f;

#define WMMA4(A, B, C) \
    __builtin_amdgcn_wmma_f32_16x16x4_f32(false, (A), false, (B), (short)0, (C), false, false)

#define NNODES 131072

// symmetric-packed, row-padded K (rows padded to multiple of 4): sum = 576
#define K_SYM 576

// workspace layout (floats)
#define OFF_SS    0                       // 576*32
#define OFF_VV    (576 * 32)              // 576*32
#define OFF_QV    (2 * 576 * 32)          // 1024*32
#define OFF_WDTS  (2 * 576 * 32 + 32768)  // 128*32
#define OFF_WDTV  (OFF_WDTS + 4096)       // 128*32

// dynamic LDS layout (floats)
#define WT_FLOATS  (2 * 576 * 32 + 32768) // 69632
#define SV_FLOATS  (4 * 16 * 128)         // 8192
#define RED_FLOATS (4 * 32 * 9)           // 1152
#define SMEM_FLOATS (WT_FLOATS + SV_FLOATS + RED_FLOATS)   // 78976 -> 315,904 B

// steps in padded symmetric row u: ceil((32-u)/4)
__device__ __host__ inline int sym_steps(int u) { return (35 - u) >> 2; }

// ---------------------------------------------------------------------------
// Prep: fold scale factors, symmetric-pack + row-pad, transpose Wd.
//   c1  = PW_DOWN*PW_QQ_S                = 1/4096 (exact)
//   c23 = c1*INV_SQRT3 = PW_DOWN*PW_QQ_V*INV_SQRT3^2 = 1/(4096*sqrt(3))
// ---------------------------------------------------------------------------
__global__ void prep_weights(const float* __restrict__ Wss, const float* __restrict__ Wvv,
                             const float* __restrict__ Wsv, const float* __restrict__ Wvs,
                             const float* __restrict__ Wds, const float* __restrict__ Wdv,
                             float* __restrict__ ws)
{
    const float c1  = 1.0f / 4096.0f;
    const float c23 = (float)(1.0 / (4096.0 * 1.7320508075688772));

    int tid = blockIdx.x * blockDim.x + threadIdx.x;
    if (tid < 32768) {
        // symmetric-packed, row-padded W_ss / W_vv
        int w = tid & 31;
        int c = (tid >> 5) & 31;      // column within padded row
        int u = tid >> 10;
        int Lp = sym_steps(u) * 4;    // padded row length
        if (c < Lp) {
            int rb = 0;               // row base (4-k steps)
            for (int j = 0; j < u; ++j) rb += sym_steps(j);
            int k = rb * 4 + c;
            float ss = 0.0f, vv = 0.0f;
            if (c == 0) {
                ss = Wss[(u * 32 + u) * 32 + w];
                vv = Wvv[(u * 32 + u) * 32 + w];
            } else if (c < 32 - u) {
                int v = u + c;
                ss = Wss[(u * 32 + v) * 32 + w] + Wss[(v * 32 + u) * 32 + w];
                vv = Wvv[(u * 32 + v) * 32 + w] + Wvv[(v * 32 + u) * 32 + w];
            }
            ws[OFF_SS + k * 32 + w] = c1 * ss;
            ws[OFF_VV + k * 32 + w] = c23 * vv;
        }
    } else if (tid < 65536) {
        // merged qv weights: Wqv[u,v,w] = c23*(W_sv[u,v,w] + W_vs[v,u,w])
        int t = tid - 32768;
        int w = t & 31;
        int v = (t >> 5) & 31;
        int u = t >> 10;
        ws[OFF_QV + ((u << 5) | v) * 32 + w] =
            c23 * (Wsv[(u * 32 + v) * 32 + w] + Wvs[(v * 32 + u) * 32 + w]);
    } else if (tid < 65536 + 8192) {
        int t = tid - 65536;
        int which = t >> 12;          // 0 -> Wd_s, 1 -> Wd_v
        int e = t & 4095;
        int f = e >> 5;
        int w = e & 31;
        const float* src = which ? Wdv : Wds;
        float* dst = ws + (which ? OFF_WDTV : OFF_WDTS);
        dst[f * 32 + w] = src[w * 128 + f];   // WdT[f][w] = Wd[w][f]
    }
}

// ---------------------------------------------------------------------------
// ss pass: acc += (s (x) s) * Wss_sym   (row-padded symmetric sweep)
// ---------------------------------------------------------------------------
__device__ __forceinline__ void gemm_ss(const float* __restrict__ svw,
                                        const float* __restrict__ wtb,
                                        int lane, v8f& acc0, v8f& acc1)
{
    const int m = lane & 15, col = lane & 15, koff = (lane >> 4) << 1;
    const float* srow = svw + m * 128;
    const float* wr = wtb;
    for (int u = 0; u < 32; ++u) {
        const float xu = srow[u];
        const int steps = sym_steps(u);
        for (int t = 0; t < steps; ++t) {
            const int c = t * 4 + koff;
            const float y0 = srow[u + c], y1 = srow[u + c + 1];
            v2f A; A.x = xu * y0; A.y = xu * y1;
            const float* w0 = wr + c * 32;
            v2f B0; B0.x = w0[col];      B0.y = w0[32 + col];
            v2f B1; B1.x = w0[col + 16]; B1.y = w0[32 + col + 16];
            acc0 = WMMA4(A, B0, acc0);
            acc1 = WMMA4(A, B1, acc1);
        }
        wr += steps * 128;
    }
}

// ---------------------------------------------------------------------------
// vv pass, all 3 components sharing B: acc += sum_i (v_i (x) v_i) * Wvv_sym
// ---------------------------------------------------------------------------
__device__ __forceinline__ void gemm_vv3(const float* __restrict__ svw,
                                         const float* __restrict__ wtb,
                                         int lane, v8f& acc0, v8f& acc1)
{
    const int m = lane & 15, col = lane & 15, koff = (lane >> 4) << 1;
    const float* srow = svw + m * 128;
    const float* wr = wtb;
    for (int u = 0; u < 32; ++u) {
        const float x0 = srow[32 + u], x1 = srow[64 + u], x2 = srow[96 + u];
        const int steps = sym_steps(u);
        for (int t = 0; t < steps; ++t) {
            const int c = t * 4 + koff;
            const int v = u + c;
            const float* w0 = wr + c * 32;
            v2f B0; B0.x = w0[col];      B0.y = w0[32 + col];
            v2f B1; B1.x = w0[col + 16]; B1.y = w0[32 + col + 16];
            v2f A0; A0.x = x0 * srow[32 + v]; A0.y = x0 * srow[32 + v + 1];
            v2f A1; A1.x = x1 * srow[64 + v]; A1.y = x1 * srow[64 + v + 1];
            v2f A2; A2.x = x2 * srow[96 + v]; A2.y = x2 * srow[96 + v + 1];
            acc0 = WMMA4(A0, B0, acc0);
            acc1 = WMMA4(A0, B1, acc1);
            acc0 = WMMA4(A1, B0, acc0);
            acc1 = WMMA4(A1, B1, acc1);
            acc0 = WMMA4(A2, B0, acc0);
            acc1 = WMMA4(A2, B1, acc1);
        }
        wr += steps * 128;
    }
}

// ---------------------------------------------------------------------------
// qv pass, all 3 components sharing B: accV_i += (s (x) v_i) * Wqv  (K=1024)
// Column index c = t*4+koff is independent of the row u, so each lane's 8
// y-pairs per component are preloaded once into registers: steady state is
// 2 ds_load_2addr (B) per 6 WMMAs, A fragments are pure v_pk_mul_f32.
// ---------------------------------------------------------------------------
__device__ __forceinline__ void gemm_qv3(const float* __restrict__ svw,
                                         const float* __restrict__ wtb,
                                         int lane,
                                         v8f& a0l, v8f& a0h, v8f& a1l, v8f& a1h,
                                         v8f& a2l, v8f& a2h)
{
    const int m = lane & 15, col = lane & 15, koff = (lane >> 4) << 1;
    const float* srow = svw + m * 128;

    v2f yx[8], yy[8], yz[8];
#pragma unroll
    for (int t = 0; t < 8; ++t) {
        const int c = t * 4 + koff;
        yx[t].x = srow[32 + c]; yx[t].y = srow[32 + c + 1];
        yy[t].x = srow[64 + c]; yy[t].y = srow[64 + c + 1];
        yz[t].x = srow[96 + c]; yz[t].y = srow[96 + c + 1];
    }

    const float* wr = wtb;
    for (int u = 0; u < 32; ++u) {
        const float xu = srow[u];
#pragma unroll
        for (int t = 0; t < 8; ++t) {
            const int c = t * 4 + koff;
            const float* w0 = wr + c * 32;
            v2f B0; B0.x = w0[col];      B0.y = w0[32 + col];
            v2f B1; B1.x = w0[col + 16]; B1.y = w0[32 + col + 16];
            v2f A0 = yx[t] * xu;
            v2f A1 = yy[t] * xu;
            v2f A2 = yz[t] * xu;
            a0l = WMMA4(A0, B0, a0l);
            a0h = WMMA4(A0, B1, a0h);
            a1l = WMMA4(A1, B0, a1l);
            a1h = WMMA4(A1, B1, a1h);
            a2l = WMMA4(A2, B0, a2l);
            a2h = WMMA4(A2, B1, a2h);
        }
        wr += 1024;                              // 32 cols * 32 w
    }
}

// Readout t_s tile: nf_s(16x128) * WdT_s(128x16), same C-layout as acc.
__device__ __forceinline__ void readout_s(const float* __restrict__ nf, int node0,
                                          const float* __restrict__ wdT,
                                          int lane, v8f& t0, v8f& t1)
{
    const int m = lane & 15, col = lane & 15, koff = (lane >> 4) << 1;
    const float* row = nf + (size_t)(node0 + m) * 512;
#pragma unroll 4
    for (int kk = 0; kk < 32; ++kk) {
        const int f0 = kk * 4 + koff;
        v2f A; A.x = row[f0]; A.y = row[f0 + 1];
        const float* w0 = wdT + f0 * 32;
        v2f B0; B0.x = w0[col];      B0.y = w0[32 + col];
        v2f B1; B1.x = w0[col + 16]; B1.y = w0[32 + col + 16];
        t0 = WMMA4(A, B0, t0);
        t1 = WMMA4(A, B1, t1);
    }
}

// Readout t_v tiles for all 3 components; A elements are 6 contiguous floats.
__device__ __forceinline__ void readout_v3(const float* __restrict__ nf, int node0,
                                           const float* __restrict__ wdT, int lane,
                                           v8f& t0l, v8f& t0h, v8f& t1l, v8f& t1h,
                                           v8f& t2l, v8f& t2h)
{
    const int m = lane & 15, col = lane & 15, koff = (lane >> 4) << 1;
    const float* row = nf + (size_t)(node0 + m) * 512 + 128;
#pragma unroll 2
    for (int kk = 0; kk < 32; ++kk) {
        const int f0 = kk * 4 + koff;
        const float* p = row + f0 * 3;           // nf_v[n, f0, 0..2], nf_v[n, f0+1, 0..2]
        const float* w0 = wdT + f0 * 32;
        v2f B0; B0.x = w0[col];      B0.y = w0[32 + col];
        v2f B1; B1.x = w0[col + 16]; B1.y = w0[32 + col + 16];
        v2f A0; A0.x = p[0]; A0.y = p[3];
        v2f A1; A1.x = p[1]; A1.y = p[4];
        v2f A2; A2.x = p[2]; A2.y = p[5];
        t0l = WMMA4(A0, B0, t0l);
        t0h = WMMA4(A0, B1, t0h);
        t1l = WMMA4(A1, B0, t1l);
        t1h = WMMA4(A1, B1, t1h);
        t2l = WMMA4(A2, B0, t2l);
        t2h = WMMA4(A2, B1, t2h);
    }
}

// ---------------------------------------------------------------------------
// Main kernel: 128 threads = 4 wave32, one 16-node M-tile per wave.
// ---------------------------------------------------------------------------
__global__ __launch_bounds__(128, 1)
void energy_kernel(const float* __restrict__ node_feats,
                   const float* __restrict__ charges0,
                   const float* __restrict__ chargesI,
                   const float* __restrict__ Was, const float* __restrict__ Wav,
                   const float* __restrict__ ba,
                   const float* __restrict__ Wbs, const float* __restrict__ Wbv,
                   const float* __restrict__ bb,
                   const float* __restrict__ ws,
                   float* __restrict__ out)
{
    extern __shared__ float smem[];
    float* wt  = smem;                          // 69632 floats
    float* sv  = smem + WT_FLOATS;              // 4 * 2048 floats
    float* red = sv + SV_FLOATS;                // 4 * 288 floats

    const int tid  = threadIdx.x;
    const int lane = tid & 31;
    const int wave = tid >> 5;

    // stage all quadratic weights into LDS (float4 copies)
    for (int i = tid; i < WT_FLOATS / 4; i += blockDim.x)
        ((float4*)wt)[i] = ((const float4*)ws)[i];
    // zero reduction scratch (padded sym reads may touch its first floats)
    for (int i = tid; i < RED_FLOATS; i += blockDim.x) red[i] = 0.0f;

    // stage s[16][32] and v[16][32][3] for this wave's 16 nodes
    const int node0 = (blockIdx.x * 4 + wave) * 16;
    float* svw = sv + wave * 2048;              // [m][128]: s | vx | vy | vz
    {
        int m = lane >> 1;
        int h = lane & 1;
        int n = node0 + m;
        float s0  = charges0[n * 4 + 0];
        float v0x = charges0[n * 4 + 1], v0y = charges0[n * 4 + 2], v0z = charges0[n * 4 + 3];
        float si  = chargesI[n * 4 + 0];
        float vix = chargesI[n * 4 + 1], viy = chargesI[n * 4 + 2], viz = chargesI[n * 4 + 3];
        for (int j = 0; j < 16; ++j) {
            int u = h * 16 + j;
            float was = Was[u], wbs = Wbs[u], wav = Wav[u], wbv = Wbv[u];
            svw[m * 128 +       u] = si * was + s0 * wbs + ba[u] + bb[u];
            svw[m * 128 +  32 + u] = vix * wav + v0x * wbv;
            svw[m * 128 +  64 + u] = viy * wav + v0y * wbv;
            svw[m * 128 +  96 + u] = viz * wav + v0z * wbv;
        }
    }
    __syncthreads();

    // quadratic contractions
    v8f accS0 = {}, accS1 = {};
    v8f aV0l = {}, aV0h = {}, aV1l = {}, aV1h = {}, aV2l = {}, aV2h = {};

    gemm_ss (svw, wt + OFF_SS, lane, accS0, accS1);
    gemm_vv3(svw, wt + OFF_VV, lane, accS0, accS1);
    gemm_qv3(svw, wt + OFF_QV, lane, aV0l, aV0h, aV1l, aV1h, aV2l, aV2h);

    // readout: t tiles share acc's C-layout -> elementwise combine
    v8f ep = {};
    {
        v8f t0 = {}, t1 = {};
        readout_s(node_feats, node0, ws + OFF_WDTS, lane, t0, t1);
        ep += accS0 * t0 + accS1 * t1;
    }
    {
        v8f t0l = {}, t0h = {}, t1l = {}, t1h = {}, t2l = {}, t2h = {};
        readout_v3(node_feats, node0, ws + OFF_WDTV, lane, t0l, t0h, t1l, t1h, t2l, t2h);
        ep += aV0l * t0l + aV0h * t0h;
        ep += aV1l * t1l + aV1h * t1h;
        ep += aV2l * t2l + aV2h * t2h;
    }

    // cross-lane reduce: element (lane c, reg r) is node {r | r+8} partial
    float* redw = red + wave * (32 * 9);
    for (int r = 0; r < 8; ++r) redw[lane * 9 + r] = ep[r];
    __syncthreads();
    if (lane < 16) {
        int n = lane;
        int r = (n < 8) ? n : (n - 8);
        int cbase = (n < 8) ? 0 : 16;
        float e = 0.0f;
        for (int c = 0; c < 16; ++c) e += redw[(cbase + c) * 9 + r];
        out[node0 + n] = e;
    }
}

// ---------------------------------------------------------------------------
extern "C" void kernel_launch(void* const* d_in, const int* in_sizes, int n_in,
                              void* d_out, int out_size, void* d_ws, size_t ws_size,
                              hipStream_t stream) {
    const float* node_feats = (const float*)d_in[0];
    const float* charges0   = (const float*)d_in[1];
    const float* chargesI   = (const float*)d_in[2];
    // d_in[3..7] (edge_feats/edge_attrs/field_feats/edge_index/batch) are unused
    const float* Was = (const float*)d_in[8];
    const float* Wav = (const float*)d_in[9];
    const float* ba  = (const float*)d_in[10];
    const float* Wbs = (const float*)d_in[11];
    const float* Wbv = (const float*)d_in[12];
    const float* bb  = (const float*)d_in[13];
    const float* Wss = (const float*)d_in[14];
    const float* Wvv = (const float*)d_in[15];
    const float* Wsv = (const float*)d_in[16];
    const float* Wvs = (const float*)d_in[17];
    const float* Wds = (const float*)d_in[18];
    const float* Wdv = (const float*)d_in[19];
    float* out = (float*)d_out;
    float* ws  = (float*)d_ws;

    // fold scales, symmetric-pack + pad, transpose Wd into workspace
    prep_weights<<<288, 256, 0, stream>>>(Wss, Wvv, Wsv, Wvs, Wds, Wdv, ws);

    // 2048 blocks x 128 threads (4 wave32 tiles of 16 nodes) = 131072 nodes
    size_t shmem = (size_t)SMEM_FLOATS * sizeof(float);   // ~316 KB of 320 KB/WGP
    energy_kernel<<<NNODES / 64, 128, shmem, stream>>>(
        node_feats, charges0, chargesI, Was, Wav, ba, Wbs, Wbv, bb, ws, out);
}